// MyLinear_73667279061712
// MI455X (gfx1250) — compile-verified
//
#include <hip/hip_runtime.h>

#define Bn 64
#define Tn 512
#define Dn 256
#define Cn 20
#define LN_EPS 1e-5f

typedef __attribute__((ext_vector_type(2))) float v2f;
typedef __attribute__((ext_vector_type(8))) float v8f;

#define XPAD 260   // padded row stride (floats) for x tile  (260 % 64 = 4 -> conflict-free strided reads)
#define WPAD 260   // padded k stride (floats) for W^T tile

// ---------------------------------------------------------------------------
// Kernel 1: embedding gather (async-to-LDS) + LayerNorm + (x @ W + b) via
// fp32 WMMA + log_softmax. One wave per 16-row tile of the (B*T, D) matrix.
// ---------------------------------------------------------------------------
__global__ __launch_bounds__(32) void fused_logits_kernel(
    const int*   __restrict__ words,
    const float* __restrict__ emb,
    const float* __restrict__ ln_g,
    const float* __restrict__ ln_b,
    const float* __restrict__ W,      // [D][C] row-major
    const float* __restrict__ bias,   // [C]
    float*       __restrict__ logits) // [B*T][C]
{
    __shared__ __align__(16) float wlds[32][WPAD]; // W^T padded to 32 cols (cols 20..31 = 0)
    __shared__ __align__(16) float xs[16][XPAD];   // 16 activation rows
    __shared__ float mu_s[16];
    __shared__ float rs_s[16];
    __shared__ float flds[16][36];

    const int lane = threadIdx.x;      // 0..31
    const int row0 = blockIdx.x * 16;  // first global row of this tile

    // ---- gather 16 embedding rows (1 KB each) straight into LDS with
    //      CDNA5 async copies (ASYNCcnt-tracked, no VGPR round-trip)
    #pragma unroll 4
    for (int r = 0; r < 16; ++r) {
        int w = words[row0 + r];
        const float* src = emb + (long)w * Dn;
        unsigned lds0 = (unsigned)(size_t)&xs[r][lane * 4];
        unsigned lds1 = (unsigned)(size_t)&xs[r][128 + lane * 4];
        const float* g0 = src + lane * 4;
        const float* g1 = src + 128 + lane * 4;
        asm volatile("global_load_async_to_lds_b128 %0, %1, off"
                     :: "v"(lds0), "v"(g0) : "memory");
        asm volatile("global_load_async_to_lds_b128 %0, %1, off"
                     :: "v"(lds1), "v"(g1) : "memory");
    }

    // ---- while the gather is in flight: stage W transposed into LDS.
    // Coalesced float4 reads of W (row-major [k][c]), scattered LDS stores.
    for (int j = lane; j < (Dn * Cn) / 4; j += 32) {   // 1280 float4s
        float4 v = ((const float4*)W)[j];
        int base = j * 4;
        wlds[(base + 0) % Cn][(base + 0) / Cn] = v.x;
        wlds[(base + 1) % Cn][(base + 1) / Cn] = v.y;
        wlds[(base + 2) % Cn][(base + 2) / Cn] = v.z;
        wlds[(base + 3) % Cn][(base + 3) / Cn] = v.w;
    }
    // zero-fill padding columns 20..31 (float4 stores)
    {
        const float4 z = {0.0f, 0.0f, 0.0f, 0.0f};
        for (int j = lane; j < (32 - Cn) * (Dn / 4); j += 32) {  // 768 float4s
            int c  = Cn + (j >> 6);       // j / 64
            int k4 = (j & 63) * 4;
            *(float4*)&wlds[c][k4] = z;
        }
    }

    // bias fragments for the epilogue (branchless clamp; cols >=20 never read)
    const int m  = lane & 15;
    const int hi = lane >> 4;
    const float bias0 = bias[m];
    const float bias1 = bias[(16 + m) < Cn ? (16 + m) : (Cn - 1)];

    asm volatile("s_wait_asynccnt 0" ::: "memory");  // gather complete
    __syncthreads();

    // ---- LayerNorm stats: 2 lanes per row, each scans 128 elements
    {
        int r = lane >> 1, h = lane & 1;
        float s = 0.0f, sq = 0.0f;
        const float* xr = &xs[r][h * 128];
        for (int i = 0; i < 128; ++i) { float v = xr[i]; s += v; sq += v * v; }
        s  += __shfl_xor(s, 1, 32);
        sq += __shfl_xor(sq, 1, 32);
        float mu   = s * (1.0f / 256.0f);
        float var  = sq * (1.0f / 256.0f) - mu * mu;
        float rstd = rsqrtf(var + LN_EPS);
        if (h == 0) { mu_s[r] = mu; rs_s[r] = rstd; }
    }
    __syncthreads();

    // ---- normalize in place with gamma/beta
    {
        int r = lane >> 1, h = lane & 1;
        float mu = mu_s[r], rstd = rs_s[r];
        float*       xr = &xs[r][h * 128];
        const float* gg = ln_g + h * 128;
        const float* bb = ln_b + h * 128;
        for (int i = 0; i < 128; ++i)
            xr[i] = (xr[i] - mu) * rstd * gg[i] + bb[i];
    }
    __syncthreads();

    // ---- WMMA: feats[16][32] = xn[16][256] @ Wpad[256][32], K in steps of 4
    // A 16x4 f32 layout: lane m (0..15): K = 4k+{0,1}; lanes 16..31: K = 4k+2+{0,1}
    // B 4x16 f32 layout mirrors A (rows striped across VGPRs, halves across lane groups)
    v8f acc0 = {0.f,0.f,0.f,0.f,0.f,0.f,0.f,0.f};
    v8f acc1 = {0.f,0.f,0.f,0.f,0.f,0.f,0.f,0.f};
    #pragma unroll 4
    for (int k = 0; k < Dn / 4; ++k) {
        int kb = 4 * k + 2 * hi;
        v2f a   = *(const v2f*)&xs[m][kb];
        v2f b0  = *(const v2f*)&wlds[m][kb];
        v2f b1  = *(const v2f*)&wlds[m + 16][kb];
        acc0 = __builtin_amdgcn_wmma_f32_16x16x4_f32(false, a, false, b0,
                                                     (short)0, acc0, false, false);
        acc1 = __builtin_amdgcn_wmma_f32_16x16x4_f32(false, a, false, b1,
                                                     (short)0, acc1, false, false);
    }

    // ---- scatter accumulators (+bias) to LDS: acc elem r -> row (r + 8*hi)
    // store all 32 cols unconditionally; cols 20..31 are never read back
    #pragma unroll
    for (int r = 0; r < 8; ++r) {
        int mm = r + 8 * hi;
        flds[mm][m]      = acc0[r] + bias0;   // cols 0..15
        flds[mm][m + 16] = acc1[r] + bias1;   // cols 16..31 (16..19 valid)
    }
    __syncthreads();

    // ---- per-row log_softmax over 20 classes; lane r handles row r
    if (lane < 16) {
        float f[Cn];
        #pragma unroll
        for (int c = 0; c < Cn; ++c) f[c] = flds[lane][c];
        // pairwise max tree (depth ~5)
        float t[Cn];
        #pragma unroll
        for (int c = 0; c < Cn; ++c) t[c] = f[c];
        #pragma unroll
        for (int st = 1; st < Cn; st *= 2)
            #pragma unroll
            for (int i = 0; i + st < Cn; i += 2 * st)
                t[i] = fmaxf(t[i], t[i + st]);
        float mx = t[0];
        float e[Cn];
        #pragma unroll
        for (int c = 0; c < Cn; ++c) e[c] = __expf(f[c] - mx);
        #pragma unroll
        for (int st = 1; st < Cn; st *= 2)
            #pragma unroll
            for (int i = 0; i + st < Cn; i += 2 * st)
                e[i] += e[i + st];
        float lse = mx + __logf(e[0]);
        float* out = logits + (long)(row0 + lane) * Cn;
        #pragma unroll
        for (int c = 0; c < Cn; ++c) out[c] = f[c] - lse;
    }
}

// ---------------------------------------------------------------------------
// Kernel 2: CRF forward (sequential over T) + gold score. One wave per batch b.
// Lane j (j<20) owns alpha_j and trans column j in registers.
// Pairwise reduction trees keep the per-step dependent chain short.
// ---------------------------------------------------------------------------
__global__ __launch_bounds__(32) void crf_kernel(
    const float* __restrict__ logits,  // [B][T][C]
    const int*   __restrict__ seq_len, // [B]
    const int*   __restrict__ target,  // [B][T]
    const float* __restrict__ trans,   // [C][C]
    const float* __restrict__ startv,  // [C]
    const float* __restrict__ endv,    // [C]
    float*       __restrict__ nll)     // [B]
{
    const int b    = blockIdx.x;
    const int lane = threadIdx.x;
    const int n    = seq_len[b];                 // 1..T
    const int*   tt = target + b * Tn;
    const float* lg = logits + (long)b * Tn * Cn;

    // ---- gold score, parallel over t, wave-reduced
    float gold = 0.0f;
    for (int t = lane; t < n; t += 32)
        gold += lg[t * Cn + tt[t]];
    for (int t = lane + 1; t < n; t += 32)
        gold += trans[tt[t - 1] * Cn + tt[t]];
    for (int off = 16; off > 0; off >>= 1)
        gold += __shfl_down(gold, off, 32);      // lane 0 holds total

    // ---- preload trans column j into registers
    float trcol[Cn];
    #pragma unroll
    for (int i = 0; i < Cn; ++i)
        trcol[i] = (lane < Cn) ? trans[i * Cn + lane] : 0.0f;

    float alpha = (lane < Cn) ? (lg[lane] + startv[lane]) : -__builtin_inff();

    // ---- forward recursion: alpha_j = logsumexp_i(alpha_i + trans[i][j]) + emit_j
    for (int t = 1; t < n; ++t) {
        float emit = (lane < Cn) ? lg[t * Cn + lane] : 0.0f;  // issued early
        float v[Cn];
        #pragma unroll
        for (int i = 0; i < Cn; ++i)
            v[i] = __shfl(alpha, i, 32) + trcol[i];   // broadcast alpha_i (parallel)
        // pairwise max tree (depth ~5 instead of serial 19)
        float tm[Cn];
        #pragma unroll
        for (int i = 0; i < Cn; ++i) tm[i] = v[i];
        #pragma unroll
        for (int st = 1; st < Cn; st *= 2)
            #pragma unroll
            for (int i = 0; i + st < Cn; i += 2 * st)
                tm[i] = fmaxf(tm[i], tm[i + st]);
        float mx = tm[0];
        // exp + pairwise sum tree
        float e[Cn];
        #pragma unroll
        for (int i = 0; i < Cn; ++i) e[i] = __expf(v[i] - mx);
        #pragma unroll
        for (int st = 1; st < Cn; st *= 2)
            #pragma unroll
            for (int i = 0; i + st < Cn; i += 2 * st)
                e[i] += e[i + st];
        float na = mx + __logf(e[0]) + emit;
        alpha = (lane < Cn) ? na : -__builtin_inff();
    }

    // ---- normalizer = logsumexp_j(alpha_j + end_j)
    float val = (lane < Cn) ? (alpha + endv[lane]) : -__builtin_inff();
    float mx = val;
    for (int off = 16; off > 0; off >>= 1)
        mx = fmaxf(mx, __shfl_xor(mx, off, 32));
    float s = (lane < Cn) ? __expf(val - mx) : 0.0f;
    for (int off = 16; off > 0; off >>= 1)
        s += __shfl_xor(s, off, 32);

    if (lane == 0) {
        float normalizer = mx + __logf(s);
        float g = gold + startv[tt[0]] + endv[tt[n - 1]];
        nll[b] = normalizer - g;
    }
}

// ---------------------------------------------------------------------------
// Kernel 3: mean over 64 per-batch NLLs
// ---------------------------------------------------------------------------
__global__ __launch_bounds__(32) void reduce_kernel(
    const float* __restrict__ nll, float* __restrict__ out)
{
    int lane = threadIdx.x;
    float v = nll[lane] + nll[lane + 32];
    for (int off = 16; off > 0; off >>= 1)
        v += __shfl_down(v, off, 32);
    if (lane == 0) out[0] = v * (1.0f / (float)Bn);
}

extern "C" void kernel_launch(void* const* d_in, const int* in_sizes, int n_in,
                              void* d_out, int out_size, void* d_ws, size_t ws_size,
                              hipStream_t stream)
{
    const int*   words  = (const int*)  d_in[0];
    const int*   seqlen = (const int*)  d_in[1];
    const int*   target = (const int*)  d_in[2];
    const float* emb    = (const float*)d_in[3];
    const float* ln_g   = (const float*)d_in[4];
    const float* ln_b   = (const float*)d_in[5];
    const float* W      = (const float*)d_in[6];
    const float* bias   = (const float*)d_in[7];
    const float* trans  = (const float*)d_in[8];
    const float* startv = (const float*)d_in[9];
    const float* endv   = (const float*)d_in[10];
    (void)in_sizes; (void)n_in; (void)out_size; (void)ws_size;

    float* logits = (float*)d_ws;                      // B*T*C floats = 2.62 MB
    float* nll    = logits + (size_t)Bn * Tn * Cn;     // B floats

    fused_logits_kernel<<<(Bn * Tn) / 16, 32, 0, stream>>>(
        words, emb, ln_g, ln_b, W, bias, logits);
    crf_kernel<<<Bn, 32, 0, stream>>>(
        logits, seqlen, target, trans, startv, endv, nll);
    reduce_kernel<<<1, 32, 0, stream>>>(nll, (float*)d_out);
}